// GraphSAGE_29016799052486
// MI455X (gfx1250) — compile-verified
//
#include <hip/hip_runtime.h>
#include <hip/hip_bf16.h>

#define N_NODES 100000
#define FEATS   128

typedef __attribute__((ext_vector_type(2))) float        v2f;
typedef __attribute__((ext_vector_type(8))) float        v8f;
typedef __attribute__((ext_vector_type(4))) unsigned int v4u;
typedef __attribute__((ext_vector_type(4))) int          v4i;
typedef __attribute__((ext_vector_type(8))) int          v8i;

// ---------------------------------------------------------------------------
// Degree: deg[dst[e]] += 1.0f  (hardware fp32 atomic)
// ---------------------------------------------------------------------------
__global__ __launch_bounds__(256) void sage_degree_kernel(
    const int* __restrict__ dst, float* __restrict__ deg, int n_edges)
{
    int e = blockIdx.x * blockDim.x + threadIdx.x;
    if (e >= n_edges) return;
    unsafeAtomicAdd(&deg[dst[e]], 1.0f);
}

// ---------------------------------------------------------------------------
// Scatter: agg[dst[e]][:] += h[src[e]][:]   one wave32 per edge,
// each lane moves a float4 (32 lanes * 4 = 128 floats).
// ---------------------------------------------------------------------------
__global__ __launch_bounds__(256) void sage_scatter_kernel(
    const float* __restrict__ h, const int* __restrict__ src,
    const int* __restrict__ dst, float* __restrict__ agg, int n_edges)
{
    int gid  = blockIdx.x * blockDim.x + threadIdx.x;
    int edge = gid >> 5;
    int lane = gid & 31;
    if (edge >= n_edges) return;
    int s = src[edge];
    int d = dst[edge];
    const float4 v = ((const float4*)(h + (size_t)s * FEATS))[lane];
    float* out = agg + (size_t)d * FEATS + lane * 4;
    unsafeAtomicAdd(out + 0, v.x);
    unsafeAtomicAdd(out + 1, v.y);
    unsafeAtomicAdd(out + 2, v.z);
    unsafeAtomicAdd(out + 3, v.w);
}

// ---------------------------------------------------------------------------
// Mean (in place): agg[n][:] /= max(deg[n], 1)
// ---------------------------------------------------------------------------
__global__ __launch_bounds__(256) void sage_mean_kernel(
    float* __restrict__ agg, const float* __restrict__ deg, int n_nodes)
{
    int gid  = blockIdx.x * blockDim.x + threadIdx.x;
    int node = gid >> 5;
    int j4   = gid & 31;
    if (node >= n_nodes) return;
    float inv = 1.0f / fmaxf(deg[node], 1.0f);
    float4* p = (float4*)(agg + (size_t)node * FEATS) + j4;
    float4  v = *p;
    v.x *= inv; v.y *= inv; v.z *= inv; v.w *= inv;
    *p = v;
}

// ---------------------------------------------------------------------------
// TDM: DMA an 80x128 f32 tile (row stride 128 floats in memory) into LDS,
// inserting 4 pad DWORDs after every 128 DWORDs -> padded LDS stride of 132
// floats (bank-conflict-free reads). Descriptor per CDNA5 ISA §8.3/8.4.
//   group0: count=1 | lds_addr | global_addr(57b) | type=2
//   group1: data_size=2(4B), pad_enable, pad_interval=6(128 DW), pad_amount=3
//           (4 DW), tensor_dim0=tile_dim0=128, tensor_dim1=tile_dim1=80,
//           tensor_dim0_stride=128
// ---------------------------------------------------------------------------
#define ROWS_PER_BLOCK 80          // 100000 % 80 == 0 -> 1250 blocks, no tail
#define ROW_TILES      5
#define LDS_STRIDE     132         // 128+4 pad: bank = 4*row + k, conflict-free

__device__ __forceinline__ void tdm_load_tile_80x128(const float* gsrc,
                                                     unsigned lds_off)
{
    unsigned long long ga = (unsigned long long)(size_t)gsrc;
    v4u g0;
    g0[0] = 1u;                                       // count=1, user-mode D#
    g0[1] = lds_off;                                  // lds_addr (bytes)
    g0[2] = (unsigned)(ga & 0xFFFFFFFFu);             // global_addr[31:0]
    g0[3] = (unsigned)((ga >> 32) & 0x01FFFFFFu)      // global_addr[56:32]
          | (2u << 30);                               // type = 2 ("image")
    v8i g1;
    g1[0] = (int)((2u << 16)      // data_size = 4 bytes
                | (1u << 20)      // pad_enable
                | (6u << 22)      // pad_interval: 128 DWORDs between pads
                | (3u << 25));    // pad_amount:   4 DWORDs of pad
    g1[1] = (int)(128u << 16);                        // tensor_dim0 = 128 (lo16)
    g1[2] = (int)(((unsigned)ROWS_PER_BLOCK) << 16);  // tensor_dim1 = 80 (lo16)
    g1[3] = (int)(128u << 16);                        // tile_dim0 = 128
    g1[4] = (int)ROWS_PER_BLOCK;                      // tile_dim1 = 80, tile_dim2 = 0
    g1[5] = (int)128;                                 // tensor_dim0_stride = 128
    g1[6] = 0;
    g1[7] = 0;
    v4i z4 = {0, 0, 0, 0};
#if defined(__clang_major__) && (__clang_major__ >= 23)
    v8i z8 = {0, 0, 0, 0, 0, 0, 0, 0};
    __builtin_amdgcn_tensor_load_to_lds(g0, g1, z4, z4, z8, 0);
#else
    __builtin_amdgcn_tensor_load_to_lds(g0, g1, z4, z4, 0);
#endif
}

// ---------------------------------------------------------------------------
// Fused SAGE GEMM:  Out = act(H @ Wself + M @ Wneigh + bias)
// Block: 256 threads = 8 waves handles an 80-row strip (5 row-tiles of 16).
// Wave w owns columns [16w, 16w+16); each weight fetch feeds 5 WMMAs.
// K=128 via v_wmma_f32_16x16x4_f32. Tiles staged by TDM (TENSORcnt).
// ---------------------------------------------------------------------------
__global__ __launch_bounds__(256) void sage_gemm_kernel(
    const float* __restrict__ H, const float* __restrict__ M,
    const float* __restrict__ Wself, const float* __restrict__ Wneigh,
    const float* __restrict__ bias, float* __restrict__ Out, int do_leaky)
{
    __shared__ float sH[ROWS_PER_BLOCK * LDS_STRIDE];
    __shared__ float sM[ROWS_PER_BLOCK * LDS_STRIDE];

    const int block_row = blockIdx.x * ROWS_PER_BLOCK;
    const int t = threadIdx.x;

    // Wave 0 issues the two tile DMAs and waits on TENSORcnt; the workgroup
    // barrier releases everyone once the LDS tiles are resident.
    if (t < 32) {
        tdm_load_tile_80x128(H + (size_t)block_row * FEATS,
                             (unsigned)(size_t)(void*)sH);
        tdm_load_tile_80x128(M + (size_t)block_row * FEATS,
                             (unsigned)(size_t)(void*)sM);
        __builtin_amdgcn_s_wait_tensorcnt(0);
    }
    __syncthreads();

    const int wave  = t >> 5;
    const int lane  = t & 31;
    const int n0    = wave * 16;
    const int arow  = lane & 15;        // A-matrix row within tile
    const int col   = lane & 15;        // B/C column
    const int khalf = (lane >> 4) * 2;  // 0 for lanes 0-15, 2 for 16-31

    v8f acc[ROW_TILES] = {};

    #pragma unroll 4
    for (int k0 = 0; k0 < FEATS; k0 += 4) {
        const int kb = k0 + khalf;
        v2f bs, bn;
        bs.x = Wself [(size_t)kb       * FEATS + n0 + col];
        bs.y = Wself [(size_t)(kb + 1) * FEATS + n0 + col];
        bn.x = Wneigh[(size_t)kb       * FEATS + n0 + col];
        bn.y = Wneigh[(size_t)(kb + 1) * FEATS + n0 + col];
        #pragma unroll
        for (int r = 0; r < ROW_TILES; ++r) {
            const float* hrow = sH + (size_t)(r * 16 + arow) * LDS_STRIDE + kb;
            const float* mrow = sM + (size_t)(r * 16 + arow) * LDS_STRIDE + kb;
            v2f a;
            a.x = hrow[0]; a.y = hrow[1];
            acc[r] = __builtin_amdgcn_wmma_f32_16x16x4_f32(
                false, a, false, bs, (short)0, acc[r], false, false);
            a.x = mrow[0]; a.y = mrow[1];
            acc[r] = __builtin_amdgcn_wmma_f32_16x16x4_f32(
                false, a, false, bn, (short)0, acc[r], false, false);
        }
    }

    const float bv = bias[n0 + col];
    #pragma unroll
    for (int r = 0; r < ROW_TILES; ++r) {
        const int mbase = block_row + r * 16 + ((lane >> 4) * 8);
        #pragma unroll
        for (int g = 0; g < 8; ++g) {
            float v = acc[r][g] + bv;
            if (do_leaky) v = (v > 0.0f) ? v : 0.2f * v;
            Out[(size_t)(mbase + g) * FEATS + n0 + col] = v;
        }
    }
}

// ---------------------------------------------------------------------------
// Host pipeline
// ---------------------------------------------------------------------------
extern "C" void kernel_launch(void* const* d_in, const int* in_sizes, int n_in,
                              void* d_out, int out_size, void* d_ws, size_t ws_size,
                              hipStream_t stream)
{
    const float* emb    = (const float*)d_in[0];
    const float* W1s    = (const float*)d_in[1];
    const float* W1n    = (const float*)d_in[2];
    const float* b1     = (const float*)d_in[3];
    const float* W2s    = (const float*)d_in[4];
    const float* W2n    = (const float*)d_in[5];
    const float* b2     = (const float*)d_in[6];
    const int*   edges  = (const int*)d_in[7];

    const int n_edges = in_sizes[7] / 2;
    const int* src = edges;
    const int* dst = edges + n_edges;

    const size_t NF = (size_t)N_NODES * FEATS;
    float* agg = (float*)d_ws;          // [N, F]
    float* h1  = agg + NF;              // [N, F]
    float* deg = h1 + NF;               // [N]
    float* out = (float*)d_out;

    const int scatter_blocks = (n_edges * 32 + 255) / 256;
    const int edge_blocks    = (n_edges + 255) / 256;
    const int mean_blocks    = (N_NODES * 32 + 255) / 256;
    const int gemm_blocks    = N_NODES / ROWS_PER_BLOCK;   // 100000/80 = 1250

    // Degree (same for both layers)
    hipMemsetAsync(deg, 0, (size_t)N_NODES * sizeof(float), stream);
    sage_degree_kernel<<<edge_blocks, 256, 0, stream>>>(dst, deg, n_edges);

    // ---- Layer 1 ----
    hipMemsetAsync(agg, 0, NF * sizeof(float), stream);
    sage_scatter_kernel<<<scatter_blocks, 256, 0, stream>>>(emb, src, dst, agg, n_edges);
    sage_mean_kernel<<<mean_blocks, 256, 0, stream>>>(agg, deg, N_NODES);
    sage_gemm_kernel<<<gemm_blocks, 256, 0, stream>>>(emb, agg, W1s, W1n, b1, h1, 1);

    // ---- Layer 2 ----
    hipMemsetAsync(agg, 0, NF * sizeof(float), stream);
    sage_scatter_kernel<<<scatter_blocks, 256, 0, stream>>>(h1, src, dst, agg, n_edges);
    sage_mean_kernel<<<mean_blocks, 256, 0, stream>>>(agg, deg, N_NODES);
    sage_gemm_kernel<<<gemm_blocks, 256, 0, stream>>>(h1, agg, W2s, W2n, b2, out, 0);
}